// FSCILGate_7473243095601
// MI455X (gfx1250) — compile-verified
//
#include <hip/hip_runtime.h>

// ---------------- problem constants (from reference) ----------------
#define N_TOKENS   32768
#define DIM        1024
#define NEXP       64
#define CAPACITY_F 40960.0f       // int(1.25 * 32768)
#define EPS        1e-6f
#define AUXW       0.01f

// ---------------- tiling ----------------
#define ROW_TILES_PER_WAVE 2                                  // B reuse factor
#define ROWS_PER_WAVE      (16 * ROW_TILES_PER_WAVE)          // 32
#define WAVES_PER_BLOCK    8
#define ROWS_PER_BLOCK     (ROWS_PER_WAVE * WAVES_PER_BLOCK)  // 256
#define NBLOCKS            (N_TOKENS / ROWS_PER_BLOCK)        // 128

typedef float v2f __attribute__((ext_vector_type(2)));
typedef float v4f __attribute__((ext_vector_type(4)));
typedef float v8f __attribute__((ext_vector_type(8)));

// =====================================================================
// Kernel 1: WMMA GEMM (x @ w_gate^T + b) -> softmax -> top-2 mask ->
//           masked scores to d_out, per-block partial column sums to ws.
// One wave computes a 32(row) x 64(expert) tile: 2 row tiles share the
// same B fragments (w_gate stays in WGP$/L2), 8 f32 WMMA accumulators.
// =====================================================================
__global__ __launch_bounds__(256, 1)
void moe_gate_main(const float* __restrict__ x,
                   const float* __restrict__ w,
                   const float* __restrict__ bias,
                   float* __restrict__ out,
                   float* __restrict__ partials)
{
    __shared__ float smem[WAVES_PER_BLOCK][3][NEXP];   // 6 KB

    const int tid  = threadIdx.x;
    const int wave = tid >> 5;
    const int lane = tid & 31;
    const int lm   = lane & 15;          // M (A) / N (B,C) index within tile
    const int half = lane >> 4;          // K-pair select for A/B, row-half for C
    const int kb   = half * 2;           // K offset {0,2}

    const int row_base = blockIdx.x * ROWS_PER_BLOCK + wave * ROWS_PER_WAVE;

    v8f acc[ROW_TILES_PER_WAVE][4];
    #pragma unroll
    for (int r = 0; r < ROW_TILES_PER_WAVE; ++r)
        #pragma unroll
        for (int t = 0; t < 4; ++t)
            acc[r][t] = (v8f){0,0,0,0,0,0,0,0};

    // A-matrix (16x4 f32): lane lm holds row, K = kb, kb+1
    const float* xrow0 = x + (size_t)(row_base + lm) * DIM;
    const float* xrow1 = x + (size_t)(row_base + 16 + lm) * DIM;
    // B-matrix (4x16 f32): lane lm holds column (16t+lm) of w_gate^T,
    // i.e. row (16t+lm) of w_gate, K = kb, kb+1
    const float* wp[4];
    #pragma unroll
    for (int t = 0; t < 4; ++t)
        wp[t] = w + (size_t)(16 * t + lm) * DIM;

    for (int k = 0; k < DIM; k += 4) {
        v2f a0 = *(const v2f*)(xrow0 + k + kb);
        v2f a1 = *(const v2f*)(xrow1 + k + kb);
        v2f b[4];
        #pragma unroll
        for (int t = 0; t < 4; ++t)
            b[t] = *(const v2f*)(wp[t] + k + kb);
        #pragma unroll
        for (int t = 0; t < 4; ++t)
            acc[0][t] = __builtin_amdgcn_wmma_f32_16x16x4_f32(
                false, a0, false, b[t], (short)0, acc[0][t], false, false);
        #pragma unroll
        for (int t = 0; t < 4; ++t)
            acc[1][t] = __builtin_amdgcn_wmma_f32_16x16x4_f32(
                false, a1, false, b[t], (short)0, acc[1][t], false, false);
    }

    // bias per column (column n = 16t + lm, constant across the 8 C rows)
    float bb[4];
    #pragma unroll
    for (int t = 0; t < 4; ++t)
        bb[t] = bias[16 * t + lm];

    // per-lane partial column sums over this wave's 32 rows
    float pm[4] = {0,0,0,0};   // masked score sum  (denominator)
    float pr[4] = {0,0,0,0};   // raw score sum     (importance)
    float pl[4] = {0,0,0,0};   // mask count        (load)

    // C layout: VGPR i holds row (i + 8*half), column lm within each tile.
    #pragma unroll
    for (int r = 0; r < ROW_TILES_PER_WAVE; ++r) {
        #pragma unroll
        for (int i = 0; i < 8; ++i) {
            const int row = row_base + 16 * r + i + half * 8;
            float v[4];
            #pragma unroll
            for (int t = 0; t < 4; ++t)
                v[t] = acc[r][t][i] + bb[t];

            // ---- softmax over 64 experts (4 regs x 16 lanes of half) ----
            float mx = fmaxf(fmaxf(v[0], v[1]), fmaxf(v[2], v[3]));
            #pragma unroll
            for (int m = 8; m >= 1; m >>= 1)
                mx = fmaxf(mx, __shfl_xor(mx, m, 32));

            float ev[4];
            float s = 0.0f;
            #pragma unroll
            for (int t = 0; t < 4; ++t) { ev[t] = __expf(v[t] - mx); s += ev[t]; }
            #pragma unroll
            for (int m = 8; m >= 1; m >>= 1)
                s += __shfl_xor(s, m, 32);
            const float inv = 1.0f / s;

            float raw[4];
            #pragma unroll
            for (int t = 0; t < 4; ++t) raw[t] = ev[t] * inv;

            // ---- top-2 by logit (same ordering as raw scores) ----
            float b1v = v[0]; int c1 = lm;
            #pragma unroll
            for (int t = 1; t < 4; ++t) {
                const int cc = 16 * t + lm;
                if (v[t] > b1v) { b1v = v[t]; c1 = cc; }
            }
            #pragma unroll
            for (int m = 8; m >= 1; m >>= 1) {
                const float ov = __shfl_xor(b1v, m, 32);
                const int   oc = __shfl_xor(c1,  m, 32);
                if (ov > b1v || (ov == b1v && oc < c1)) { b1v = ov; c1 = oc; }
            }
            float b2v = -3.4e38f; int c2 = -1;
            #pragma unroll
            for (int t = 0; t < 4; ++t) {
                const int cc = 16 * t + lm;
                if (cc != c1 && v[t] > b2v) { b2v = v[t]; c2 = cc; }
            }
            #pragma unroll
            for (int m = 8; m >= 1; m >>= 1) {
                const float ov = __shfl_xor(b2v, m, 32);
                const int   oc = __shfl_xor(c2,  m, 32);
                if (ov > b2v || (ov == b2v && oc < c2)) { b2v = ov; c2 = oc; }
            }

            // ---- masked scores out + partial sums ----
            #pragma unroll
            for (int t = 0; t < 4; ++t) {
                const int  cc   = 16 * t + lm;
                const bool keep = (cc == c1) || (cc == c2);
                const float mv  = keep ? raw[t] : 0.0f;
                out[(size_t)row * NEXP + cc] = mv;
                pm[t] += mv;
                pr[t] += raw[t];
                pl[t] += keep ? 1.0f : 0.0f;
            }
        }
    }

    // combine the two 16-lane halves (row halves of each tile)
    #pragma unroll
    for (int t = 0; t < 4; ++t) {
        pm[t] += __shfl_xor(pm[t], 16, 32);
        pr[t] += __shfl_xor(pr[t], 16, 32);
        pl[t] += __shfl_xor(pl[t], 16, 32);
    }
    if (half == 0) {
        #pragma unroll
        for (int t = 0; t < 4; ++t) {
            smem[wave][0][16 * t + lm] = pm[t];
            smem[wave][1][16 * t + lm] = pr[t];
            smem[wave][2][16 * t + lm] = pl[t];
        }
    }
    __syncthreads();

    // fixed-order cross-wave reduction -> per-block partials (deterministic)
    if (tid < NEXP) {
        #pragma unroll
        for (int stat = 0; stat < 3; ++stat) {
            float s = 0.0f;
            #pragma unroll
            for (int wv = 0; wv < WAVES_PER_BLOCK; ++wv)
                s += smem[wv][stat][tid];
            partials[((size_t)blockIdx.x * 3 + stat) * NEXP + tid] = s;
        }
    }
}

// =====================================================================
// Kernel 2: reduce 128 block partials -> per-expert scale + aux loss
// =====================================================================
__global__ __launch_bounds__(64, 1)
void moe_gate_reduce(const float* __restrict__ partials,
                     float* __restrict__ scale,
                     float* __restrict__ aux_out)
{
    __shared__ float sh[2];
    const int e = threadIdx.x;   // 0..63

    float dsum = 0.0f, isum = 0.0f, lsum = 0.0f;
    for (int b = 0; b < NBLOCKS; ++b) {
        const float* p = partials + (size_t)b * 3 * NEXP;
        dsum += p[e];
        isum += p[NEXP + e];
        lsum += p[2 * NEXP + e];
    }
    scale[e] = CAPACITY_F / (dsum + EPS);

    // aux = AUXW * mean_e(importance_e * load_e) * E^2
    float prod = (isum * (1.0f / N_TOKENS)) * (lsum * (1.0f / N_TOKENS));
    #pragma unroll
    for (int m = 16; m >= 1; m >>= 1)
        prod += __shfl_xor(prod, m, 32);
    if ((threadIdx.x & 31) == 0) sh[threadIdx.x >> 5] = prod;
    __syncthreads();
    if (threadIdx.x == 0) {
        const float tot = sh[0] + sh[1];
        aux_out[0] = AUXW * (tot / (float)NEXP) * (float)(NEXP * NEXP);
    }
}

// =====================================================================
// Kernel 3: in-place rescale of masked scores by per-expert scale
// =====================================================================
__global__ __launch_bounds__(256, 1)
void moe_gate_scale(float* __restrict__ out, const float* __restrict__ scale)
{
    const size_t idx = ((size_t)blockIdx.x * blockDim.x + threadIdx.x) * 4;
    v4f v = *(const v4f*)(out + idx);
    const v4f s = *(const v4f*)(scale + (idx & 63));   // e = flat & 63, 16B aligned
    v *= s;
    *(v4f*)(out + idx) = v;
}

// =====================================================================
extern "C" void kernel_launch(void* const* d_in, const int* in_sizes, int n_in,
                              void* d_out, int out_size, void* d_ws, size_t ws_size,
                              hipStream_t stream)
{
    const float* x    = (const float*)d_in[0];   // [32768, 1024]
    const float* w    = (const float*)d_in[1];   // [64, 1024]
    const float* bias = (const float*)d_in[2];   // [64]
    float* out = (float*)d_out;                  // [32768*64] scores + [1] aux

    float* partials = (float*)d_ws;                          // 128*3*64 floats
    float* scale    = partials + (size_t)NBLOCKS * 3 * NEXP; // 64 floats

    moe_gate_main<<<NBLOCKS, 256, 0, stream>>>(x, w, bias, out, partials);
    moe_gate_reduce<<<1, 64, 0, stream>>>(partials, scale,
                                          out + (size_t)N_TOKENS * NEXP);
    const int total_v4 = (N_TOKENS * NEXP) / 4;              // 524288
    moe_gate_scale<<<total_v4 / 256, 256, 0, stream>>>(out, scale);
}